// TinyMambaSTFTUNetV2_41154376630318
// MI455X (gfx1250) — compile-verified
//
#include <hip/hip_runtime.h>
#include <math.h>

typedef float v2f __attribute__((ext_vector_type(2)));
typedef float v8f __attribute__((ext_vector_type(8)));

// ---------------- model dims ----------------
#define BB    4
#define TLEN  5000
#define NFFT  256
#define HOP   64
#define NBIN  129
#define NF    79        // frames
#define F1    129
#define F2    65
#define F3    33
#define TT    79
#define LM    (F3*TT)   // 2607 mamba sequence length
#define FT1   (F1*TT)   // 10191
#define FT2   (F2*TT)   // 5135
#define FT3   (F3*TT)   // 2607
#define MROW  (BB*NF)   // 316 rows in (i)DFT GEMMs
#define DINN  128
#define DSTT  64

// ---------------- GEMM flags ----------------
#define GF_TA       1
#define GF_TB       2
#define GF_BIAS_COL 4
#define GF_BIAS_ROW 8
#define GF_TANH     16

#define TWO_PI_OVER_N (6.2831853071795864f/256.0f)

// =====================================================================
// Generic batched f32 WMMA GEMM: C = act(A*B + bias (+ accin))
// Flags are a compile-time template parameter so trans/bias/activation
// resolve statically.  One 16x16 tile per wave via V_WMMA_F32_16X16X4_F32;
// tile id is wave-uniform, so the early return and the fast/edge-path
// branch keep EXEC all-ones at every wmma issue.  Interior tiles run a
// guard-free pointer-stepped K loop (unrolled 4x); only edge tiles and
// the K%4 tail pay per-element guards.
// =====================================================================
template<int FLAGS>
__global__ void k_gemm_t(const float* __restrict__ A, const float* __restrict__ Bm,
                         float* __restrict__ C, const float* __restrict__ bias,
                         const float* __restrict__ accin,
                         int M, int N, int K, int lda, int ldb, int ldc,
                         long sA, long sB, long sC)
{
    int lane = threadIdx.x & 31;
    int wave = threadIdx.x >> 5;
    int tiles_n = (N + 15) >> 4;
    int tiles_m = (M + 15) >> 4;
    int tile = blockIdx.x * (blockDim.x >> 5) + wave;
    if (tile >= tiles_m * tiles_n) return;           // wave-uniform exit
    int m0 = (tile / tiles_n) << 4;
    int n0 = (tile % tiles_n) << 4;

    const float* Ab = A  + (long)blockIdx.y * sA;
    const float* Bb = Bm + (long)blockIdx.y * sB;
    float*       Cb = C  + (long)blockIdx.y * sC;
    const float* Rb = accin ? (accin + (long)blockIdx.y * sC) : (const float*)0;

    int row = m0 + (lane & 15);
    int col = n0 + (lane & 15);
    int kb  = (lane >> 4) << 1;                      // 0 or 2
    int K4  = K & ~3;

    v8f acc = {};

    if (m0 + 16 <= M && n0 + 16 <= N) {
        // ---------- fast path: full tile, no per-lane guards ----------
        const float* pA; long stepA, offA1;
        if (FLAGS & GF_TA) { pA = Ab + (long)kb * lda + row; stepA = 4L * lda; offA1 = lda; }
        else               { pA = Ab + (long)row * lda + kb; stepA = 4;        offA1 = 1;   }
        const float* pB; long stepB, offB1;
        if (FLAGS & GF_TB) { pB = Bb + (long)col * ldb + kb; stepB = 4;        offB1 = 1;   }
        else               { pB = Bb + (long)kb * ldb + col; stepB = 4L * ldb; offB1 = ldb; }

        int k0 = 0;
        #pragma unroll 4
        for (; k0 < K4; k0 += 4) {
            v2f a, b;
            a[0] = pA[0]; a[1] = pA[offA1];
            b[0] = pB[0]; b[1] = pB[offB1];
            acc = __builtin_amdgcn_wmma_f32_16x16x4_f32(false, a, false, b,
                                                        (short)0, acc, false, false);
            pA += stepA; pB += stepB;
        }
        if (k0 < K) {                                // K tail (<=3 extra cols)
            int k = k0 + kb;
            float a0 = 0.f, a1 = 0.f, b0 = 0.f, b1 = 0.f;
            if (k < K) {
                a0 = (FLAGS & GF_TA) ? Ab[(long)k * lda + row] : Ab[(long)row * lda + k];
                b0 = (FLAGS & GF_TB) ? Bb[(long)col * ldb + k] : Bb[(long)k * ldb + col];
            }
            if (k + 1 < K) {
                a1 = (FLAGS & GF_TA) ? Ab[(long)(k + 1) * lda + row] : Ab[(long)row * lda + k + 1];
                b1 = (FLAGS & GF_TB) ? Bb[(long)col * ldb + k + 1]   : Bb[(long)(k + 1) * ldb + col];
            }
            v2f a, b;
            a[0] = a0; a[1] = a1; b[0] = b0; b[1] = b1;
            acc = __builtin_amdgcn_wmma_f32_16x16x4_f32(false, a, false, b,
                                                        (short)0, acc, false, false);
        }
    } else {
        // ---------- edge path: per-element guards ----------
        bool rok = row < M;
        bool cok = col < N;
        for (int k0 = 0; k0 < K; k0 += 4) {
            int k = k0 + kb;
            float a0 = 0.f, a1 = 0.f, b0 = 0.f, b1 = 0.f;
            if (rok && k     < K) a0 = (FLAGS & GF_TA) ? Ab[(long)k    *lda + row] : Ab[(long)row*lda + k    ];
            if (rok && (k+1) < K) a1 = (FLAGS & GF_TA) ? Ab[(long)(k+1)*lda + row] : Ab[(long)row*lda + k + 1];
            if (cok && k     < K) b0 = (FLAGS & GF_TB) ? Bb[(long)col*ldb + k    ] : Bb[(long)k    *ldb + col];
            if (cok && (k+1) < K) b1 = (FLAGS & GF_TB) ? Bb[(long)col*ldb + k + 1] : Bb[(long)(k+1)*ldb + col];
            v2f a, b;
            a[0] = a0; a[1] = a1; b[0] = b0; b[1] = b1;
            acc = __builtin_amdgcn_wmma_f32_16x16x4_f32(false, a, false, b,
                                                        (short)0, acc, false, false);
        }
    }

    if (col < N) {
        int rbase = m0 + ((lane >> 4) << 3);         // M = r + 8*(lane/16)
        #pragma unroll
        for (int r = 0; r < 8; ++r) {
            int rr = rbase + r;
            if (rr < M) {
                float v = acc[r];
                if (FLAGS & GF_BIAS_COL) v += bias[col];
                if (FLAGS & GF_BIAS_ROW) v += bias[rr];
                if (Rb) v += Rb[(long)rr * ldc + col];
                if (FLAGS & GF_TANH) v = tanhf(v);
                Cb[(long)rr * ldc + col] = v;
            }
        }
    }
}

// ================= STFT / iSTFT pieces =================
__global__ void k_dft_basis(float* __restrict__ Bm) {  // [256 t, 258 cols]
    int idx = blockIdx.x * blockDim.x + threadIdx.x;
    if (idx >= 256 * 258) return;
    int col = idx % 258, t = idx / 258, k = col >> 1;
    float ang = TWO_PI_OVER_N * (float)(k * t);
    Bm[idx] = (col & 1) ? -__sinf(ang) : __cosf(ang);
}
__global__ void k_idft_basis(float* __restrict__ Bm) { // [258 rows, 256 t]
    int idx = blockIdx.x * blockDim.x + threadIdx.x;
    if (idx >= 258 * 256) return;
    int t = idx % 256, rw = idx / 256, k = rw >> 1;
    float alpha = (k == 0 || k == 128) ? 1.0f : 2.0f;
    float ang = TWO_PI_OVER_N * (float)(k * t);
    Bm[idx] = ((rw & 1) ? -alpha * __sinf(ang) : alpha * __cosf(ang)) * (1.0f / 256.0f);
}
__global__ void k_frames(const float* __restrict__ x, float* __restrict__ fr) {
    int idx = blockIdx.x * blockDim.x + threadIdx.x;
    if (idx >= MROW * NFFT) return;
    int j = idx % NFFT; int m = idx / NFFT;
    int i = m % NF, b = m / NF;
    int p = i * HOP + j - NFFT / 2;                 // reflect pad
    if (p < 0) p = -p;
    if (p >= TLEN) p = 2 * TLEN - 2 - p;
    float w = 0.5f - 0.5f * __cosf(TWO_PI_OVER_N * (float)j);
    fr[idx] = x[b * TLEN + p] * w;
}
__global__ void k_feats(const float* __restrict__ X, float* __restrict__ feats) {
    int idx = blockIdx.x * blockDim.x + threadIdx.x;
    if (idx >= BB * NBIN * NF) return;
    int i = idx % NF; int k = (idx / NF) % NBIN; int b = idx / (NBIN * NF);
    const float re = X[((long)(b * NF + i)) * 258 + 2 * k];
    const float im = X[((long)(b * NF + i)) * 258 + 2 * k + 1];
    float mag = sqrtf(re * re + im * im);
    float c = 1.f, s = 0.f;
    if (mag > 0.f) { c = re / mag; s = im / mag; }
    long base = ((long)b * 3) * FT1 + (long)k * TT + i;
    feats[base]            = powf(mag + 1e-5f, 0.3f);
    feats[base + FT1]      = c;
    feats[base + 2 * FT1]  = s;
}
__global__ void k_cmul(const float* __restrict__ mask, const float* __restrict__ X,
                       float* __restrict__ Y) {
    int idx = blockIdx.x * blockDim.x + threadIdx.x;
    if (idx >= BB * NBIN * NF) return;
    int i = idx % NF; int k = (idx / NF) % NBIN; int b = idx / (NBIN * NF);
    float mr = mask[((long)b * 2)     * FT1 + (long)k * TT + i];
    float mi = mask[((long)b * 2 + 1) * FT1 + (long)k * TT + i];
    long row = ((long)(b * NF + i)) * 258;
    float re = X[row + 2 * k], im = X[row + 2 * k + 1];
    Y[row + 2 * k]     = mr * re - mi * im;
    Y[row + 2 * k + 1] = mr * im + mi * re;
}
__global__ void k_ola(const float* __restrict__ fr, float* __restrict__ out) {
    int idx = blockIdx.x * blockDim.x + threadIdx.x;
    if (idx >= BB * TLEN) return;
    int t = idx % TLEN, b = idx / TLEN;
    int tp = t + NFFT / 2;
    int s0 = tp - (NFFT - 1);
    int i0 = (s0 <= 0) ? 0 : ((s0 + HOP - 1) >> 6);
    int i1 = tp >> 6; if (i1 > NF - 1) i1 = NF - 1;
    float acc = 0.f, env = 0.f;
    for (int i = i0; i <= i1; ++i) {
        int j = tp - i * HOP;
        float w = 0.5f - 0.5f * __cosf(TWO_PI_OVER_N * (float)j);
        acc += fr[((long)(b * NF + i)) * NFFT + j] * w;
        env += w * w;
    }
    out[idx] = acc / (env > 1e-11f ? env : 1.0f);
}

// ================= conv / norm pieces =================
__global__ void k_conv3x3(const float* __restrict__ in, const float* __restrict__ w,
                          const float* __restrict__ bias, float* __restrict__ out,
                          int B, int Cin, int Cout, int Fin, int Tin,
                          int Fout, int Tout, int strideF, int dilF)
{
    long idx = (long)blockIdx.x * blockDim.x + threadIdx.x;
    long total = (long)B * Cout * Fout * Tout;
    if (idx >= total) return;
    int to = (int)(idx % Tout); long r = idx / Tout;
    int fo = (int)(r % Fout);   r /= Fout;
    int co = (int)(r % Cout);   int b = (int)(r / Cout);
    float s = bias[co];
    int fbase = fo * strideF - dilF;
    int tbase = to - 1;
    const float* wco = w + (long)co * Cin * 9;
    const float* inb = in + (long)b * Cin * Fin * Tin;
    for (int ci = 0; ci < Cin; ++ci) {
        const float* ic = inb + (long)ci * Fin * Tin;
        const float* wc = wco + ci * 9;
        #pragma unroll
        for (int kf = 0; kf < 3; ++kf) {
            int fi = fbase + kf * dilF;
            if (fi < 0 || fi >= Fin) continue;
            const float* rowp = ic + (long)fi * Tin;
            #pragma unroll
            for (int kt = 0; kt < 3; ++kt) {
                int ti = tbase + kt;
                if (ti < 0 || ti >= Tin) continue;
                s += rowp[ti] * wc[kf * 3 + kt];
            }
        }
    }
    out[idx] = s;
}

__global__ void k_inorm_prelu(float* __restrict__ x, const float* __restrict__ g,
                              const float* __restrict__ be, const float* __restrict__ a,
                              int C, int FT)
{
    __shared__ float s1[256], s2[256];
    float* p = x + (long)blockIdx.x * FT;
    float sum = 0.f, sq = 0.f;
    for (int i = threadIdx.x; i < FT; i += blockDim.x) { float v = p[i]; sum += v; sq += v * v; }
    s1[threadIdx.x] = sum; s2[threadIdx.x] = sq;
    __syncthreads();
    for (int st = 128; st > 0; st >>= 1) {
        if ((int)threadIdx.x < st) { s1[threadIdx.x] += s1[threadIdx.x + st]; s2[threadIdx.x] += s2[threadIdx.x + st]; }
        __syncthreads();
    }
    float m = s1[0] / (float)FT;
    float v = s2[0] / (float)FT - m * m;
    float inv = rsqrtf(v + 1e-5f);
    int c = blockIdx.x % C;
    float gg = g[c], bb = be[c], aa = a[c];
    for (int i = threadIdx.x; i < FT; i += blockDim.x) {
        float t = (p[i] - m) * inv * gg + bb;
        p[i] = (t >= 0.f) ? t : aa * t;
    }
}

__global__ void k_convT2x1(const float* __restrict__ in, const float* __restrict__ w,
                           const float* __restrict__ bias, float* __restrict__ out,
                           int B, int C, int Fin, int T, int Fout)
{
    long idx = (long)blockIdx.x * blockDim.x + threadIdx.x;
    long total = (long)B * C * Fout * T;
    if (idx >= total) return;
    int t = (int)(idx % T); long r = idx / T;
    int fo = (int)(r % Fout); r /= Fout;
    int co = (int)(r % C);    int b = (int)(r / C);
    int kh = fo & 1, fi = fo >> 1;
    float s = bias[co];
    if (fi < Fin) {
        for (int ci = 0; ci < C; ++ci)
            s += in[(((long)b * C + ci) * Fin + fi) * T + t] * w[((long)ci * C + co) * 2 + kh];
    }
    out[idx] = s;
}

__global__ void k_copych(const float* __restrict__ src, float* __restrict__ dst,
                         int B, int Csrc, long FT, int Cdst, int c0)
{
    long idx = (long)blockIdx.x * blockDim.x + threadIdx.x;
    long total = (long)B * Csrc * FT;
    if (idx >= total) return;
    long i = idx % FT; long r = idx / FT;
    int c = (int)(r % Csrc); int b = (int)(r / Csrc);
    dst[((long)b * Cdst + c0 + c) * FT + i] = src[idx];
}

// ================= mamba pieces =================
__global__ void k_layernorm(const float* __restrict__ x, float* __restrict__ y,
                            const float* __restrict__ g, const float* __restrict__ b_, int BL)
{
    int i = blockIdx.x * blockDim.x + threadIdx.x;
    if (i >= BL) return;
    const float* p = x + (long)i * 64; float* q = y + (long)i * 64;
    float m = 0.f;
    for (int c = 0; c < 64; ++c) m += p[c];
    m *= (1.0f / 64.0f);
    float v = 0.f;
    for (int c = 0; c < 64; ++c) { float d = p[c] - m; v += d * d; }
    v *= (1.0f / 64.0f);
    float inv = rsqrtf(v + 1e-5f);
    for (int c = 0; c < 64; ++c) q[c] = (p[c] - m) * inv * g[c] + b_[c];
}

__global__ void k_dwconv_silu(const float* __restrict__ xz, const float* __restrict__ cw,
                              const float* __restrict__ cb, float* __restrict__ si, int L)
{
    long idx = (long)blockIdx.x * blockDim.x + threadIdx.x;
    long total = (long)BB * L * DINN;
    if (idx >= total) return;
    int d = (int)(idx % DINN); long r = idx / DINN;
    int l = (int)(r % L); int b = (int)(r / L);
    const float* base = xz + (long)b * L * 256;
    float s = cb[d];
    #pragma unroll
    for (int j = 0; j < 4; ++j) {
        int ls = l - 3 + j;
        if (ls >= 0) s += base[(long)ls * 256 + d] * cw[d * 4 + j];
    }
    float sig = 1.f / (1.f + __expf(-s));
    si[idx] = s * sig;
}

__global__ void k_delta(const float* __restrict__ dbc, const float* __restrict__ dtw,
                        const float* __restrict__ dtb, float* __restrict__ del, int L)
{
    long idx = (long)blockIdx.x * blockDim.x + threadIdx.x;
    long total = (long)BB * L * DINN;
    if (idx >= total) return;
    int d = (int)(idx % DINN); long r = idx / DINN;
    int l = (int)(r % L); int b = (int)(r / L);
    const float* row = dbc + ((long)b * L + l) * 132;
    float s = dtb[d];
    #pragma unroll
    for (int rr = 0; rr < 4; ++rr) s += row[rr] * dtw[d * 4 + rr];
    del[idx] = (s > 20.f) ? s : log1pf(__expf(s));
}

// Selective scan: one block per batch, thread = d_inner channel.
// 64 states + 64 A-coeffs in registers; B/C staged through LDS per step.
__global__ void k_scan(const float* __restrict__ delta, const float* __restrict__ dbc,
                       const float* __restrict__ si, const float* __restrict__ A_log,
                       float* __restrict__ ys, int L)
{
    __shared__ float Bs[DSTT];
    __shared__ float Cs[DSTT];
    int b = blockIdx.x;
    int d = threadIdx.x;
    float h[DSTT], Ar[DSTT];
    #pragma unroll
    for (int n = 0; n < DSTT; ++n) { h[n] = 0.f; Ar[n] = -__expf(A_log[d * DSTT + n]); }
    const float* dl = delta + (long)b * L * DINN;
    const float* sl = si    + (long)b * L * DINN;
    const float* eb = dbc   + (long)b * L * 132;
    float*       yb = ys    + (long)b * L * DINN;
    for (int l = 0; l < L; ++l) {
        if (d < DSTT) Bs[d] = eb[(long)l * 132 + 4 + d];
        else          Cs[d - DSTT] = eb[(long)l * 132 + 68 + (d - DSTT)];
        __syncthreads();
        float dt = dl[(long)l * DINN + d];
        float dx = dt * sl[(long)l * DINN + d];
        float y = 0.f;
        #pragma unroll
        for (int n = 0; n < DSTT; ++n) {
            h[n] = h[n] * __expf(dt * Ar[n]) + dx * Bs[n];
            y += h[n] * Cs[n];
        }
        yb[(long)l * DINN + d] = y;
        __syncthreads();
    }
}

__global__ void k_gate(const float* __restrict__ ys, const float* __restrict__ si,
                       const float* __restrict__ D, const float* __restrict__ xz,
                       float* __restrict__ yg, int L)
{
    long idx = (long)blockIdx.x * blockDim.x + threadIdx.x;
    long total = (long)BB * L * DINN;
    if (idx >= total) return;
    int d = (int)(idx % DINN); long r = idx / DINN;
    int l = (int)(r % L); int b = (int)(r / L);
    float z = xz[((long)b * L + l) * 256 + 128 + d];
    float zs = z / (1.f + __expf(-z));
    yg[idx] = (ys[idx] + si[idx] * D[d]) * zs;
}

// =====================================================================
// Host side
// =====================================================================
#define PP(i) ((const float*)din[(i)])

static inline int gblk(long total, int thr) { return (int)((total + thr - 1) / thr); }

static void gemm(hipStream_t s, const float* A, const float* Bm, float* C,
                 const float* bias, const float* acc,
                 int M, int N, int K, int lda, int ldb, int ldc,
                 long sA, long sB, long sC, int batch, int flags)
{
    int tiles = ((M + 15) / 16) * ((N + 15) / 16);
    dim3 g((tiles + 3) / 4, batch);
    switch (flags) {
    case 0:
        k_gemm_t<0><<<g, 128, 0, s>>>(A, Bm, C, bias, acc, M, N, K, lda, ldb, ldc, sA, sB, sC); break;
    case GF_TB:
        k_gemm_t<GF_TB><<<g, 128, 0, s>>>(A, Bm, C, bias, acc, M, N, K, lda, ldb, ldc, sA, sB, sC); break;
    case GF_TA | GF_TB | GF_BIAS_COL:
        k_gemm_t<GF_TA | GF_TB | GF_BIAS_COL><<<g, 128, 0, s>>>(A, Bm, C, bias, acc, M, N, K, lda, ldb, ldc, sA, sB, sC); break;
    case GF_TB | GF_BIAS_ROW:
        k_gemm_t<GF_TB | GF_BIAS_ROW><<<g, 128, 0, s>>>(A, Bm, C, bias, acc, M, N, K, lda, ldb, ldc, sA, sB, sC); break;
    case GF_BIAS_ROW | GF_TANH:
        k_gemm_t<GF_BIAS_ROW | GF_TANH><<<g, 128, 0, s>>>(A, Bm, C, bias, acc, M, N, K, lda, ldb, ldc, sA, sB, sC); break;
    default: break; // unreachable: only the above combos are used
    }
}
static void conv3(hipStream_t s, const float* in, const float* w, const float* bias, float* out,
                  int Cin, int Cout, int Fin, int Fout, int strideF, int dilF)
{
    long total = (long)BB * Cout * Fout * TT;
    k_conv3x3<<<gblk(total, 256), 256, 0, s>>>(in, w, bias, out, BB, Cin, Cout,
                                               Fin, TT, Fout, TT, strideF, dilF);
}
static void inorm(hipStream_t s, float* x, const float* g, const float* be, const float* a,
                  int C, int FT)
{
    k_inorm_prelu<<<BB * C, 256, 0, s>>>(x, g, be, a, C, FT);
}
// conv_block: i0 -> sorted cb keys: a1,a2,b1,b2,be1,be2,g1,g2,w1,w2
static void run_cb(hipStream_t s, void* const* din, int i0,
                   const float* in, float* t, float* out,
                   int Cin, int Cout, int Fin, int Fout, int strideF)
{
    conv3(s, in, PP(i0 + 8), PP(i0 + 2), t, Cin, Cout, Fin, Fout, strideF, 1);
    inorm(s, t, PP(i0 + 6), PP(i0 + 4), PP(i0 + 0), Cout, Fout * TT);
    conv3(s, t, PP(i0 + 9), PP(i0 + 3), out, Cout, Cout, Fout, Fout, 1, 1);
    inorm(s, out, PP(i0 + 7), PP(i0 + 5), PP(i0 + 1), Cout, Fout * TT);
}

// -------- workspace arena (float offsets) --------
static const long O_DFTB  = 0;
static const long O_IDFTB = O_DFTB  + 256L * 258;
static const long O_FR    = O_IDFTB + 258L * 256;
static const long O_X     = O_FR    + (long)MROW * 256;
static const long O_Y     = O_X     + (long)MROW * 258;
static const long O_FEATS = O_Y     + (long)MROW * 258;
static const long O_MASK  = O_FEATS + (long)BB * 3 * FT1;
static const long O_S1    = O_MASK  + (long)BB * 2 * FT1;
static const long O_S2    = O_S1    + (long)BB * 32 * FT1;
static const long O_H3    = O_S2    + (long)BB * 64 * FT2;
static const long O_T     = O_H3    + (long)BB * 128 * FT3;
static const long TSLOT   = 4200000;
static const long O_TA = O_T, O_TB = O_T + TSLOT, O_TC = O_T + 2 * TSLOT;
// mamba arena aliases the temp slots (live ranges do not overlap)
static const long O_HS  = O_T;
static const long O_XN  = O_HS  + (long)BB * LM * 64;
static const long O_XZ  = O_XN  + (long)BB * LM * 64;
static const long O_SI  = O_XZ  + (long)BB * LM * 256;
static const long O_DBC = O_SI  + (long)BB * LM * 128;
static const long O_DEL = O_DBC + (long)BB * LM * 132;
static const long O_YS  = O_DEL + (long)BB * LM * 128;
static const long O_YG  = O_YS  + (long)BB * LM * 128;

// Parameter index map.  Assumption: top-level dict insertion order ('x' then
// 'params'), params flattened jax-style (dict keys sorted, lists in order).
//  0: x
//  1: bridge_b   2: bridge_w
//  3..12 : enc1 cb  (a1,a2,b1,b2,be1,be2,g1,g2,w1,w2)
// 13..22 : enc2 cb ; 23..27 enc2.dense[0] (a,b,be,g,w) ; 28..32 dense[1]
// 33..42 : enc3 cb ; 43..47 enc3.dense[0] ; 48..52 dense[1]
// 53: head_b  54: head_w
// 55+11*i : mamba[i] = A_log,D,conv_b,conv_w,dt_b,dt_w,in_proj_w,ln_b,ln_g,out_proj_w,x_proj_w
// 88: unbridge_b 89: unbridge_w
// 90..99 up1 cb ; 100 up1.up_b ; 101 up1.up_w
// 102..111 up2 cb ; 112 up2.up_b ; 113 up2.up_w

extern "C" void kernel_launch(void* const* d_in, const int* in_sizes, int n_in,
                              void* d_out, int out_size, void* d_ws, size_t ws_size,
                              hipStream_t stream)
{
    (void)in_sizes; (void)n_in; (void)out_size; (void)ws_size;
    void* const* din = d_in;
    float* W = (float*)d_ws;
    const float* x = PP(0);

    float* FR = W + O_FR;  float* Xb = W + O_X;   float* Yb = W + O_Y;
    float* FE = W + O_FEATS; float* MK = W + O_MASK;
    float* S1 = W + O_S1;  float* S2 = W + O_S2;  float* H3 = W + O_H3;
    float* TA = W + O_TA;  float* TB = W + O_TB;  float* TC = W + O_TC;

    // ---- STFT as WMMA GEMM ----
    k_dft_basis <<<gblk(256L * 258, 256), 256, 0, stream>>>(W + O_DFTB);
    k_idft_basis<<<gblk(258L * 256, 256), 256, 0, stream>>>(W + O_IDFTB);
    k_frames    <<<gblk((long)MROW * NFFT, 256), 256, 0, stream>>>(x, FR);
    gemm(stream, FR, W + O_DFTB, Xb, 0, 0, MROW, 258, 256, 256, 258, 258, 0, 0, 0, 1, 0);
    k_feats<<<gblk((long)BB * NBIN * NF, 256), 256, 0, stream>>>(Xb, FE);

    // ---- encoder ----
    run_cb(stream, din, 3,  FE, TA, S1, 3, 32, F1, F1, 1);                 // enc1 -> s1
    run_cb(stream, din, 13, S1, TA, TB, 32, 64, F1, F2, 2);                // enc2 cb
    conv3(stream, TB, PP(27), PP(24), TC, 64, 64, F2, F2, 1, 1);           // dense l0
    inorm(stream, TC, PP(26), PP(25), PP(23), 64, FT2);
    k_copych<<<gblk((long)BB * 64 * FT2, 256), 256, 0, stream>>>(TB, TA, BB, 64, FT2, 128, 0);
    k_copych<<<gblk((long)BB * 64 * FT2, 256), 256, 0, stream>>>(TC, TA, BB, 64, FT2, 128, 64);
    conv3(stream, TA, PP(32), PP(29), S2, 128, 64, F2, F2, 1, 2);          // dense l1 (dil 2)
    inorm(stream, S2, PP(31), PP(30), PP(28), 64, FT2);                    // -> s2

    run_cb(stream, din, 33, S2, TA, TB, 64, 128, F2, F3, 2);               // enc3 cb
    conv3(stream, TB, PP(47), PP(44), TC, 128, 128, F3, F3, 1, 1);
    inorm(stream, TC, PP(46), PP(45), PP(43), 128, FT3);
    k_copych<<<gblk((long)BB * 128 * FT3, 256), 256, 0, stream>>>(TB, TA, BB, 128, FT3, 256, 0);
    k_copych<<<gblk((long)BB * 128 * FT3, 256), 256, 0, stream>>>(TC, TA, BB, 128, FT3, 256, 128);
    conv3(stream, TA, PP(52), PP(49), H3, 256, 128, F3, F3, 1, 2);
    inorm(stream, H3, PP(51), PP(50), PP(48), 128, FT3);                   // -> h

    // ---- bridge (1x1 conv as GEMM): hs[b,l,c] = sum_k h3[b,k,l] * Wb[c,k] + bb[c]
    gemm(stream, H3, PP(2), W + O_HS, PP(1), 0, LM, 64, 128,
         FT3, 128, 64, 128L * FT3, 0, (long)LM * 64, BB, GF_TA | GF_TB | GF_BIAS_COL);

    // ---- mamba x3 ----
    float* HS = W + O_HS; float* XN = W + O_XN; float* XZ = W + O_XZ;
    float* SI = W + O_SI; float* DB = W + O_DBC; float* DL = W + O_DEL;
    float* YS = W + O_YS; float* YG = W + O_YG;
    for (int i = 0; i < 3; ++i) {
        int q = 55 + 11 * i;  // A_log,D,conv_b,conv_w,dt_b,dt_w,in_proj_w,ln_b,ln_g,out_proj_w,x_proj_w
        k_layernorm<<<gblk((long)BB * LM, 256), 256, 0, stream>>>(HS, XN, PP(q + 8), PP(q + 7), BB * LM);
        gemm(stream, XN, PP(q + 6), XZ, 0, 0, LM, 256, 64,
             64, 64, 256, (long)LM * 64, 0, (long)LM * 256, BB, GF_TB);
        k_dwconv_silu<<<gblk((long)BB * LM * DINN, 256), 256, 0, stream>>>(XZ, PP(q + 3), PP(q + 2), SI, LM);
        gemm(stream, SI, PP(q + 10), DB, 0, 0, LM, 132, 128,
             128, 128, 132, (long)LM * 128, 0, (long)LM * 132, BB, GF_TB);
        k_delta<<<gblk((long)BB * LM * DINN, 256), 256, 0, stream>>>(DB, PP(q + 5), PP(q + 4), DL, LM);
        k_scan<<<BB, DINN, 0, stream>>>(DL, DB, SI, PP(q + 0), YS, LM);
        k_gate<<<gblk((long)BB * LM * DINN, 256), 256, 0, stream>>>(YS, SI, PP(q + 1), XZ, YG, LM);
        gemm(stream, YG, PP(q + 9), HS, 0, HS, LM, 64, 128,           // + residual
             128, 128, 64, (long)LM * 128, 0, (long)LM * 64, BB, GF_TB);
    }

    // ---- unbridge: h3[b,co,l] = sum_c Wu[co,c]*hs[b,l,c] + bu[co]
    gemm(stream, PP(89), HS, H3, PP(88), 0, 128, LM, 64,
         64, 64, FT3, 0, (long)LM * 64, 128L * FT3, BB, GF_TB | GF_BIAS_ROW);

    // ---- up2 ----
    k_convT2x1<<<gblk((long)BB * 128 * F2 * TT, 256), 256, 0, stream>>>(H3, PP(113), PP(112), TA, BB, 128, F3, TT, F2);
    k_copych<<<gblk((long)BB * 128 * FT2, 256), 256, 0, stream>>>(TA, TB, BB, 128, FT2, 192, 0);
    k_copych<<<gblk((long)BB * 64  * FT2, 256), 256, 0, stream>>>(S2, TB, BB, 64,  FT2, 192, 128);
    run_cb(stream, din, 102, TB, TC, TA, 192, 64, F2, F2, 1);              // -> TA

    // ---- up1 ----
    k_convT2x1<<<gblk((long)BB * 64 * F1 * TT, 256), 256, 0, stream>>>(TA, PP(101), PP(100), TB, BB, 64, F2, TT, F1);
    k_copych<<<gblk((long)BB * 64 * FT1, 256), 256, 0, stream>>>(TB, TC, BB, 64, FT1, 96, 0);
    k_copych<<<gblk((long)BB * 32 * FT1, 256), 256, 0, stream>>>(S1, TC, BB, 32, FT1, 96, 64);
    run_cb(stream, din, 90, TC, TA, TB, 96, 32, F1, F1, 1);                // -> TB

    // ---- head (1x1 conv + tanh) ----
    gemm(stream, PP(54), TB, MK, PP(53), 0, 2, FT1, 32,
         32, FT1, FT1, 0, 32L * FT1, 2L * FT1, BB, GF_BIAS_ROW | GF_TANH);

    // ---- mask * X, iSTFT ----
    k_cmul<<<gblk((long)BB * NBIN * NF, 256), 256, 0, stream>>>(MK, Xb, Yb);
    gemm(stream, Yb, W + O_IDFTB, FR, 0, 0, MROW, 256, 258, 258, 256, 256, 0, 0, 0, 1, 0);
    k_ola<<<gblk((long)BB * TLEN, 256), 256, 0, stream>>>(FR, (float*)d_out);
}